// InfoNCELoss_55516747268533
// MI455X (gfx1250) — compile-verified
//
#include <hip/hip_runtime.h>
#include <hip/hip_bf16.h>
#include <math.h>

typedef __attribute__((ext_vector_type(2))) float v2f;
typedef __attribute__((ext_vector_type(8))) float v8f;

#define DIMS 24
#define INV_2SIG2 (1.0f / 1.125f)   // 2*0.75^2 = 1.125
#define INV_TAU 20.0f

// ---------------- Pass A: init workspace ----------------
__global__ void k_init(float* denom, unsigned* maxbits, float* loss_sum,
                       float* pos_cnt, int N) {
    int i = blockIdx.x * blockDim.x + threadIdx.x;
    if (i < N) denom[i] = 0.0f;
    if (i == 0) { *maxbits = 0u; *loss_sum = 0.0f; *pos_cnt = 0.0f; }
}

// ---------------- Pass B: per-pair logits via f32 WMMA ----------------
// One wave handles 16 pairs. For each K-chunk of 4 (6 chunks over D=24):
//   lane L (<16) holds diff[pair L][k0 .. k0+1]  (A vgpr0/1, B vgpr0/1 identical)
//   lane L+16    holds diff[pair L][k0+2..k0+3]
// c += A * B  ->  diag(c)[i] = sum_k diff_i[k]^2
__global__ void k_pairs_wmma(const float* __restrict__ x,
                             const int* __restrict__ pp,
                             float* __restrict__ logits,
                             unsigned* __restrict__ maxbits,
                             int P) {
    int tid  = blockIdx.x * blockDim.x + threadIdx.x;
    int wave = tid >> 5;
    int lane = threadIdx.x & 31;

    long pairBase = (long)wave * 16;
    int  slot = lane & 15;
    long p  = pairBase + slot;
    long pc = (p < P) ? p : (long)P - 1;      // clamp: keep EXEC full for WMMA

    int a = pp[pc];
    int b = pp[(long)P + pc];
    int koff = (lane >> 4) << 1;              // 0 for lanes 0-15, 2 for 16-31

    const float* xa = x + (long)a * DIMS + koff;
    const float* xb = x + (long)b * DIMS + koff;

    v8f c = {};
#pragma unroll
    for (int k0 = 0; k0 < DIMS; k0 += 4) {
        float2 va = *(const float2*)(xa + k0);   // 8B aligned: row stride 96B, offset mult of 8
        float2 vb = *(const float2*)(xb + k0);
        v2f d;
        d.x = va.x - vb.x;
        d.y = va.y - vb.y;
        // D = A*B + C, A = diff tile (16x4), B = diff^T (4x16): same lane data.
        c = __builtin_amdgcn_wmma_f32_16x16x4_f32(
                /*neg_a=*/false, d, /*neg_b=*/false, d,
                /*c_mod=*/(short)0, c, /*reuse_a=*/false, /*reuse_b=*/false);
    }

    // diag(i,i): i in 0..7 -> lane i,    vgpr i
    //            i in 8..15 -> lane i+16, vgpr i-8
    bool hasDiag = (lane < 8) || (lane >= 24);
    int  dslot   = (lane < 8) ? lane : (lane - 16);
    int  ci      = lane & 7;
    float sq = c[0];
    if (ci == 1) sq = c[1];
    if (ci == 2) sq = c[2];
    if (ci == 3) sq = c[3];
    if (ci == 4) sq = c[4];
    if (ci == 5) sq = c[5];
    if (ci == 6) sq = c[6];
    if (ci == 7) sq = c[7];

    float lmax = 0.0f;                        // logits > 0, so 0 is a neutral element
    long  pd = pairBase + dslot;
    if (hasDiag && pd < P) {
        float l2    = sqrtf(fmaxf(sq, 0.0f));
        float sim   = expf(-l2 * INV_2SIG2);
        float logit = sim * INV_TAU;
        logits[pd]  = logit;
        lmax = logit;
    }
    // wave32 max-reduce, one atomic per wave (uint order == float order for >=0)
#pragma unroll
    for (int m = 16; m > 0; m >>= 1)
        lmax = fmaxf(lmax, __shfl_xor(lmax, m, 32));
    if (lane == 0) atomicMax(maxbits, __float_as_uint(lmax));
}

// ---------------- Pass C: scatter neg exp-sims into denom ----------------
__global__ void k_scatter(const int* __restrict__ pp,
                          const int* __restrict__ cid,
                          const float* __restrict__ logits,
                          const unsigned* __restrict__ maxbits,
                          float* __restrict__ denom, int P) {
    long p = (long)blockIdx.x * blockDim.x + threadIdx.x;
    if (p >= P) return;
    float maxv = __uint_as_float(*maxbits);
    float es   = expf(logits[p] - maxv);
    int a = pp[p];
    int b = pp[(long)P + p];
    int ca = cid[a], cb = cid[b];
    bool pos = (ca == cb) && (ca != -1) && (cb != -1);
    if (!pos) atomicAdd(&denom[a], es);
}

// ---------------- Pass D: positive-pair loss reduction ----------------
__global__ void k_loss(const int* __restrict__ pp,
                       const int* __restrict__ cid,
                       const float* __restrict__ logits,
                       const unsigned* __restrict__ maxbits,
                       const float* __restrict__ denom,
                       float* __restrict__ loss_sum,
                       float* __restrict__ pos_cnt, int P) {
    __shared__ float sL[8], sC[8];
    long p = (long)blockIdx.x * blockDim.x + threadIdx.x;
    float loss = 0.0f, cnt = 0.0f;
    if (p < P) {
        int a = pp[p];
        int b = pp[(long)P + p];
        int ca = cid[a], cb = cid[b];
        if ((ca == cb) && (ca != -1) && (cb != -1)) {
            float maxv = __uint_as_float(*maxbits);
            float es   = expf(logits[p] - maxv);
            float dn   = fmaxf(denom[a], 0.0f);
            loss = log1pf(dn / es);           // == -log(num/(num+denom))
            cnt  = 1.0f;
        }
    }
#pragma unroll
    for (int m = 16; m > 0; m >>= 1) {
        loss += __shfl_xor(loss, m, 32);
        cnt  += __shfl_xor(cnt,  m, 32);
    }
    int lane = threadIdx.x & 31;
    int wv   = threadIdx.x >> 5;
    if (lane == 0) { sL[wv] = loss; sC[wv] = cnt; }
    __syncthreads();
    if (threadIdx.x == 0) {
        float tl = 0.0f, tc = 0.0f;
        int nw = blockDim.x >> 5;
        for (int i = 0; i < nw; ++i) { tl += sL[i]; tc += sC[i]; }
        if (tc != 0.0f || tl != 0.0f) {
            atomicAdd(loss_sum, tl);
            atomicAdd(pos_cnt,  tc);
        }
    }
}

// ---------------- Pass E: finalize ----------------
__global__ void k_final(const float* loss_sum, const float* pos_cnt, float* out) {
    out[0] = loss_sum[0] / pos_cnt[0];
}

extern "C" void kernel_launch(void* const* d_in, const int* in_sizes, int n_in,
                              void* d_out, int out_size, void* d_ws, size_t ws_size,
                              hipStream_t stream) {
    const float* x   = (const float*)d_in[0];   // [N, 24]
    const int*   pp  = (const int*)  d_in[1];   // [2, P]
    const int*   cid = (const int*)  d_in[2];   // [N]
    // d_in[3] (recons), d_in[4] (pts) unused by the forward.

    int N = in_sizes[2];
    int P = in_sizes[1] / 2;

    // Workspace layout (floats): [0]=max bits, [1]=loss_sum, [2]=pos_cnt,
    // [8 .. 8+N) = denom, [8+N .. 8+N+P) = logits   (~8.8 MB total)
    float*    wsf      = (float*)d_ws;
    unsigned* maxbits  = (unsigned*)wsf;
    float*    loss_sum = wsf + 1;
    float*    pos_cnt  = wsf + 2;
    float*    denom    = wsf + 8;
    float*    logits   = wsf + 8 + N;

    // A: init
    k_init<<<(N + 255) / 256, 256, 0, stream>>>(denom, maxbits, loss_sum, pos_cnt, N);

    // B: WMMA pair pass (16 pairs/wave, 8 waves/block -> 128 pairs/block)
    int pairsPerBlock = (256 / 32) * 16;
    int nbB = (P + pairsPerBlock - 1) / pairsPerBlock;
    k_pairs_wmma<<<nbB, 256, 0, stream>>>(x, pp, logits, maxbits, P);

    // C: scatter denom
    k_scatter<<<(P + 255) / 256, 256, 0, stream>>>(pp, cid, logits, maxbits, denom, P);

    // D: loss reduction
    k_loss<<<(P + 255) / 256, 256, 0, stream>>>(pp, cid, logits, maxbits, denom,
                                                loss_sum, pos_cnt, P);

    // E: finalize
    k_final<<<1, 1, 0, stream>>>(loss_sum, pos_cnt, (float*)d_out);
}